// OpticallyThinDiffraction_6279242187486
// MI455X (gfx1250) — compile-verified
//
#include <hip/hip_runtime.h>

// ---------------------------------------------------------------------------
// Ptychography forward: out[n,0,h,w] = probe[h,w] * (C_sy · P_n · C_sx^T)[h,w]
// Fourier sub-pixel shift expressed as circulant (periodic-sinc) matmuls so
// the entire op runs on the CDNA5 WMMA pipe in full fp32 (complex64 math).
// One workgroup (256 thr = 8 waves) per scan position; 2048 workgroups.
// Patch gather uses GLOBAL_LOAD_ASYNC_TO_LDS_B32 (ASYNCcnt) when available,
// overlapped with the circulant build; per-lane LDS addresses deinterleave
// the complex data into re/im planes for free.
// ---------------------------------------------------------------------------

#define PITCH 132                 // 128 + 4 floats: conflict-free column loads
#define PLANE (128 * PITCH)       // floats per LDS plane

typedef float v2f __attribute__((ext_vector_type(2)));
typedef float v8f __attribute__((ext_vector_type(8)));

#if __has_builtin(__builtin_amdgcn_global_load_async_to_lds_b32)
#define HAVE_ASYNC_LDS 1
#else
#define HAVE_ASYNC_LDS 0
#endif

#if HAVE_ASYNC_LDS
// clang prints addrspace(1) as "__device__" and addrspace(3) as "__shared__"
// in HIP diagnostics; the builtin takes (AS1 int*, AS3 int*, imm, imm).
typedef __attribute__((address_space(1))) int* gas_ptr;
typedef __attribute__((address_space(3))) int* las_ptr;
#endif

__device__ __forceinline__ void wait_async_zero() {
#if __has_builtin(__builtin_amdgcn_s_wait_asynccnt)
  __builtin_amdgcn_s_wait_asynccnt(0);
#else
  asm volatile("s_wait_asynccnt 0" ::: "memory");
#endif
}

__device__ __forceinline__ v8f wmma_f32(v2f a, v2f b, v8f c) {
  // D(16x16,f32) = A(16x4,f32) * B(4x16,f32) + C
  return __builtin_amdgcn_wmma_f32_16x16x4_f32(
      false, a, false, b, (short)0, c, false, false);
}

__global__ void __launch_bounds__(256)
ptycho_shift_wmma(const float2* __restrict__ psi,   // 1024x1024 complex64
                  const float*  __restrict__ scan,  // 2048x2 float32
                  const float2* __restrict__ kern,  // 1x1x128x128 complex64
                  float2*       __restrict__ out)   // 2048x1x128x128 complex64
{
  extern __shared__ float smem[];
  float*  Ar  = smem;                  // phase1: circulant C_sy (re); then T(re)
  float*  Ai  = Ar + PLANE;
  float*  Br  = Ai + PLANE;            // K-major: Br[col*PITCH + k]
  float*  Bi  = Br + PLANE;
  float2* shv = (float2*)(Bi + PLANE); // [0..127]=S_y gen, [128..255]=S_x gen

  const int n    = blockIdx.x;
  const int t    = threadIdx.x;
  const int lane = t & 31;
  const int wv   = t >> 5;       // wave 0..7
  const int lo   = lane & 15;
  const int hi   = lane >> 4;
  const int m0   = wv * 16;      // this wave's output row stripe

  const float scy = scan[2 * n + 0];
  const float scx = scan[2 * n + 1];
  const int   iy  = (int)scy;    // positive -> trunc == floor (matches ref)
  const int   ix  = (int)scx;

  // ---- kick off patch gather into B planes, K-major (transposed) ----------
  // B[k=h][col=w] stored col-major; async path deinterleaves re/im via
  // per-lane LDS destination addresses and is tracked with ASYNCcnt.
#if HAVE_ASYNC_LDS
  {
    const float* psf = (const float*)psi;
    for (int e = t; e < 128 * 128; e += 256) {
      const int h = e >> 7, w = e & 127;
      const float* g = psf + 2 * ((iy + h) * 1024 + (ix + w));
      __builtin_amdgcn_global_load_async_to_lds_b32(
          (gas_ptr)(g + 0), (las_ptr)&Br[w * PITCH + h], 0, 0);
      __builtin_amdgcn_global_load_async_to_lds_b32(
          (gas_ptr)(g + 1), (las_ptr)&Bi[w * PITCH + h], 0, 0);
    }
  }
#else
  for (int e = t; e < 128 * 128; e += 256) {
    const int h = e >> 7, w = e & 127;
    const float2 v = psi[(iy + h) * 1024 + (ix + w)];
    Br[w * PITCH + h] = v.x;
    Bi[w * PITCH + h] = v.y;
  }
#endif

  // ---- shift generator: S[d] for both axes (one sincos per thread) --------
  {
    const int   dim = t >> 7;                       // 0 -> y, 1 -> x
    const int   d   = t & 127;
    const float s   = dim ? ((float)ix - scx) : ((float)iy - scy); // (-1,0]
    float re, im;
    if (fabsf(s) < 1e-6f) {                         // integer scan: delta
      re = (d == 0) ? 1.f : 0.f; im = 0.f;
    } else {
      const float PI  = 3.14159265358979323846f;
      const float sp  = sinf(PI * s);
      const float ang = PI * ((float)d - s) * (1.0f / 128.0f);   // in (0, pi)
      const float sa  = sinf(ang);
      const float ca  = cosf(ang);
      const float par = (d & 1) ? -1.f : 1.f;                    // (-1)^d
      const float amp = -par * sp / (128.0f * sa);
      re = amp * ca;
      im = -amp * sa;
    }
    shv[t] = make_float2(re, im);
  }
  __syncthreads();                      // shv visible to all threads

  // ---- build circulant A[m][l] = S_y[(m-l) mod 128] -----------------------
  // (overlaps with the in-flight async patch DMA)
  for (int e = t; e < 128 * 128; e += 256) {
    const int m = e >> 7, l = e & 127;
    const float2 v = shv[(m - l) & 127];
    Ar[m * PITCH + l] = v.x;
    Ai[m * PITCH + l] = v.y;
  }
#if HAVE_ASYNC_LDS
  wait_async_zero();                    // drain this wave's async LDS writes
#endif
  __syncthreads();                      // A + B planes ready

  const int arow = (m0 + lo) * PITCH;   // A fragment row base for this lane
  v8f accr[8], acci[8];
  const v8f vzero = {};

  // ======================= phase 1: T = C_sy * P ===========================
#pragma unroll
  for (int jt = 0; jt < 8; ++jt) { accr[jt] = vzero; acci[jt] = vzero; }

  for (int kb = 0; kb < 128; kb += 4) {
    const int k0 = kb + 2 * hi;         // A layout: VGPR0={K0,K2}, VGPR1={K1,K3}
    const v2f ar = *(const v2f*)&Ar[arow + k0];
    const v2f ai = *(const v2f*)&Ai[arow + k0];
#pragma unroll
    for (int jt = 0; jt < 8; ++jt) {
      const int bcol = (jt * 16 + lo) * PITCH;
      const v2f br  = *(const v2f*)&Br[bcol + k0];
      const v2f bi  = *(const v2f*)&Bi[bcol + k0];
      const v2f nbi = -bi;
      accr[jt] = wmma_f32(ar, br,  accr[jt]);  // re:  ar*br - ai*bi
      accr[jt] = wmma_f32(ai, nbi, accr[jt]);
      acci[jt] = wmma_f32(ar, bi,  acci[jt]);  // im:  ar*bi + ai*br
      acci[jt] = wmma_f32(ai, br,  acci[jt]);
    }
  }

  // write T into this wave's OWN A-stripe (no other wave touches these rows)
#pragma unroll
  for (int jt = 0; jt < 8; ++jt) {
#pragma unroll
    for (int r = 0; r < 8; ++r) {
      const int m = m0 + r + 8 * hi;    // C layout: VGPR r -> rows r, r+8
      const int c = jt * 16 + lo;
      Ar[m * PITCH + c] = accr[jt][r];
      Ai[m * PITCH + c] = acci[jt][r];
    }
  }
  __syncthreads();                      // everyone done reading P from B

  // ---- rebuild B planes as C_sx^T, K-major: B[c][j] = S_x[(c-j) mod 128] --
  for (int e = t; e < 128 * 128; e += 256) {
    const int c = e >> 7, j = e & 127;
    const float2 v = shv[128 + ((c - j) & 127)];
    Br[c * PITCH + j] = v.x;
    Bi[c * PITCH + j] = v.y;
  }
  __syncthreads();

  // ======================= phase 2: OUT = T * C_sx^T =======================
#pragma unroll
  for (int jt = 0; jt < 8; ++jt) { accr[jt] = vzero; acci[jt] = vzero; }

  for (int kb = 0; kb < 128; kb += 4) {
    const int k0 = kb + 2 * hi;
    const v2f ar = *(const v2f*)&Ar[arow + k0];
    const v2f ai = *(const v2f*)&Ai[arow + k0];
#pragma unroll
    for (int jt = 0; jt < 8; ++jt) {
      const int bcol = (jt * 16 + lo) * PITCH;
      const v2f br  = *(const v2f*)&Br[bcol + k0];
      const v2f bi  = *(const v2f*)&Bi[bcol + k0];
      const v2f nbi = -bi;
      accr[jt] = wmma_f32(ar, br,  accr[jt]);
      accr[jt] = wmma_f32(ai, nbi, accr[jt]);
      acci[jt] = wmma_f32(ar, bi,  acci[jt]);
      acci[jt] = wmma_f32(ai, br,  acci[jt]);
    }
  }

  // ---- epilogue: multiply by probe, store complex64 (coalesced b64) -------
#pragma unroll
  for (int jt = 0; jt < 8; ++jt) {
#pragma unroll
    for (int r = 0; r < 8; ++r) {
      const int m = m0 + r + 8 * hi;
      const int c = jt * 16 + lo;
      const float2 kv = kern[m * 128 + c];
      const float tr = accr[jt][r], ti = acci[jt][r];
      out[(n * 128 + m) * 128 + c] =
          make_float2(kv.x * tr - kv.y * ti, kv.x * ti + kv.y * tr);
    }
  }
}

extern "C" void kernel_launch(void* const* d_in, const int* in_sizes, int n_in,
                              void* d_out, int out_size, void* d_ws, size_t ws_size,
                              hipStream_t stream) {
  const float2* psi  = (const float2*)d_in[0];   // complex64 1024x1024
  const float*  scan = (const float*)d_in[1];    // float32  2048x2
  const float2* kern = (const float2*)d_in[2];   // complex64 1x1x128x128
  float2*       out  = (float2*)d_out;           // complex64 2048x1x128x128

  const int nscan = in_sizes[1] / 2;
  const size_t shmem = (size_t)(4 * PLANE) * sizeof(float) + 256 * sizeof(float2);

  (void)hipFuncSetAttribute(reinterpret_cast<const void*>(ptycho_shift_wmma),
                            hipFuncAttributeMaxDynamicSharedMemorySize, (int)shmem);
  ptycho_shift_wmma<<<nscan, 256, shmem, stream>>>(psi, scan, kern, out);
}